// ContinuousLocationMapXX_62139586839055
// MI455X (gfx1250) — compile-verified
//
#include <hip/hip_runtime.h>

// ContinuousLocationMap: B=32 samples, 512x512x3 f32 map each.
// out[b,i,j] = (0.633, j*d, i*d) except inside the 8x8 window of the LAST
// location covering (i,j), where out = (0, loc.x, loc.y).
// d = 10/512 = 0.01953125 (exact in f32).

#define BINSN   512
#define WIN     8
#define HALF    4
#define LOCS    64
#define DELTA_F 0.01953125f
#define CORRV   0.633f

// Native clang vector type: accepted by __builtin_nontemporal_store and
// lowers to one global_store_b128 (with TH=NT cache-control bits).
typedef float v4f __attribute__((ext_vector_type(4)));

// ---------------- Phase 1: streaming base fill (bandwidth bound) -------------
// Each thread writes 4 consecutive pixels (12 floats = 48B) of one row as three
// 16B-aligned nontemporal b128 stores. Pattern is independent of batch index.
__global__ void __launch_bounds__(256)
clm_fill_base(float* __restrict__ out) {
    unsigned t = blockIdx.x * 256u + threadIdx.x;
    unsigned P = t * 4u;                 // first pixel id (flat over B*512*512)
    unsigned j = P & (BINSN - 1u);       // column (x-channel value = j*d)
    unsigned i = (P >> 9) & (BINSN - 1u);// row    (y-channel value = i*d)

    float y  = (float)i * DELTA_F;
    float x0 = (float)j * DELTA_F;
    float x1 = x0 + DELTA_F;
    float x2 = x1 + DELTA_F;
    float x3 = x2 + DELTA_F;

    v4f f0 = {CORRV, x0, y, CORRV};
    v4f f1 = {x1, y, CORRV, x2};
    v4f f2 = {y, CORRV, x3, y};

    v4f* p = (v4f*)(out + (size_t)P * 3u);  // 48B-aligned => 16B-aligned
    __builtin_nontemporal_store(f0, p + 0);
    __builtin_nontemporal_store(f1, p + 1);
    __builtin_nontemporal_store(f2, p + 2);
}

// ---------------- Phase 2: winner-resolved window scatter --------------------
// One block per sample. 64 locations x 64 window pixels = 4096 writes/sample.
// last-write-wins resolved in parallel: pixel of location l is written only if
// no later location l' > l covers it (coverage = clamped [-4,+3] box).
__global__ void __launch_bounds__(256)
clm_scatter(const float* __restrict__ locs, float* __restrict__ out) {
    __shared__ int   sx[LOCS], sy[LOCS];
    __shared__ float slx[LOCS], sly[LOCS];

    const int b = blockIdx.x;
    const int t = threadIdx.x;

    if (t < LOCS) {
        float lx = locs[(b * LOCS + t) * 2 + 0];
        float ly = locs[(b * LOCS + t) * 2 + 1];
        slx[t] = lx;
        sly[t] = ly;
        sx[t] = (int)(lx / DELTA_F);   // IEEE divide, trunc-toward-zero: matches
        sy[t] = (int)(ly / DELTA_F);   // (loc/delta).astype(int32) for loc >= 0
    }
    __syncthreads();

    float* outb = out + (size_t)b * (BINSN * BINSN * 3);

    for (int g = t; g < LOCS * WIN * WIN; g += 256) {
        const int l  = g >> 6;             // location index 0..63
        const int p  = g & 63;             // window pixel 0..63
        const int ox = (p >> 3) - HALF;    // -4..3
        const int oy = (p & 7)  - HALF;    // -4..3

        const int px = min(BINSN - 1, max(0, sx[l] + ox));
        const int py = min(BINSN - 1, max(0, sy[l] + oy));

        // Skip if any LATER location's (clamped) window covers this pixel.
        bool covered = false;
        for (int m = l + 1; m < LOCS; ++m) {
            const int lo0 = min(BINSN - 1, max(0, sx[m] - HALF));
            const int hi0 = min(BINSN - 1, max(0, sx[m] + HALF - 1));
            const int lo1 = min(BINSN - 1, max(0, sy[m] - HALF));
            const int hi1 = min(BINSN - 1, max(0, sy[m] + HALF - 1));
            if ((px >= lo0) & (px <= hi0) & (py >= lo1) & (py <= hi1)) {
                covered = true;
                break;
            }
        }
        if (!covered) {
            float* q = outb + ((size_t)px * BINSN + py) * 3;
            q[0] = 0.0f;
            q[1] = slx[l];
            q[2] = sly[l];
        }
    }
}

extern "C" void kernel_launch(void* const* d_in, const int* in_sizes, int n_in,
                              void* d_out, int out_size, void* d_ws, size_t ws_size,
                              hipStream_t stream) {
    const float* locations = (const float*)d_in[0];   // [B, 64, 2] f32
    float* out = (float*)d_out;                       // [B, 512, 512, 3] f32

    const int B = in_sizes[0] / (LOCS * 2);           // 32

    // Phase 1: B*512*512 pixels, 4 pixels/thread, 256 threads/block.
    const int fill_blocks = B * (BINSN * BINSN / 4 / 256);   // B*256
    clm_fill_base<<<fill_blocks, 256, 0, stream>>>(out);

    // Phase 2: one block per sample; same-stream launch orders it after fill.
    clm_scatter<<<B, 256, 0, stream>>>(locations, out);
}